// Processor_11845519802432
// MI455X (gfx1250) — compile-verified
//
#include <hip/hip_runtime.h>
#include <hip/hip_bf16.h>

#define HID 128
#define LDS_STRIDE 132   // pad 128->132 so 16x4 A-fragment reads spread across banks
#define KSTEPS 32        // 128 / 4

typedef float v2f __attribute__((ext_vector_type(2)));
typedef float v8f __attribute__((ext_vector_type(8)));

// monotone float <-> uint mapping so unsigned atomicMax == float max
__device__ __forceinline__ unsigned fkey_enc(float f) {
    unsigned u = __float_as_uint(f);
    return (u & 0x80000000u) ? ~u : (u | 0x80000000u);
}
__device__ __forceinline__ float fkey_dec(unsigned k) {
    unsigned u = (k & 0x80000000u) ? (k ^ 0x80000000u) : ~k;
    return __uint_as_float(u);
}

__global__ __launch_bounds__(256) void zero_u32(unsigned* __restrict__ p, int n) {
    int t = blockIdx.x * blockDim.x + threadIdx.x;
    if (t < n) p[t] = 0u;
}

// Repack 128x128 B (row-major, K x N) into WMMA fragment order:
// Bp[((kstep*128 + n)*2 + half)] = float2{ B[4k+2h][n], B[4k+2h+1][n] }
// -> in the GEMM loop each lane does ONE aligned global_load_b64 per k-step.
__global__ __launch_bounds__(256) void pack_b(const float* __restrict__ B,
                                              float* __restrict__ Bp) {
    int t = blockIdx.x * blockDim.x + threadIdx.x;   // one thread per (kstep, n)
    if (t >= KSTEPS * HID) return;
    int kstep = t >> 7, n = t & 127;
    float2 lo = make_float2(B[(4 * kstep + 0) * HID + n], B[(4 * kstep + 1) * HID + n]);
    float2 hi = make_float2(B[(4 * kstep + 2) * HID + n], B[(4 * kstep + 3) * HID + n]);
    float2* dst = (float2*)Bp + (size_t)t * 2;
    dst[0] = lo;
    dst[1] = hi;
}

// C[M x 128] = act( A1 @ B1  (+ A2 @ B2)  + bias ),  B given packed.
// block = 256 threads = 8 waves; block covers 32 rows x 128 cols.
// wave w owns columns [16w,16w+16) for BOTH 16-row tiles -> each B fragment
// feeds two WMMAs.
__global__ __launch_bounds__(256) void gemm32(
    const float* __restrict__ A1, const float* __restrict__ B1p,
    const float* __restrict__ A2, const float* __restrict__ B2p,
    const float* __restrict__ bias, float* __restrict__ C, int relu, int M)
{
    __shared__ float As[2][2][16 * LDS_STRIDE];   // [A-set][m-tile][...]
    const int tid  = threadIdx.x;
    const int wave = tid >> 5;
    const int lane = tid & 31;
    const int half = lane >> 4;   // 0: K=k0..k0+1, 1: K=k0+2..k0+3 (ISA 16x4 A layout)
    const int lrow = lane & 15;
    const int m0 = blockIdx.x * 32;
    const int n0 = wave * 16;
    const bool tile1_ok = (m0 + 16) < M;

    // cooperative stage of the two 16x128 A tiles (per A-set) into LDS
    {
        int r  = tid >> 4;
        int c  = (tid & 15) * 8;
        int r1 = tile1_ok ? (m0 + 16 + r) : (m0 + r);   // clamp OOB tile to valid rows
        const float4* sa = (const float4*)(A1 + (size_t)(m0 + r) * HID + c);
        const float4* sb = (const float4*)(A1 + (size_t)r1 * HID + c);
        *(float4*)&As[0][0][r * LDS_STRIDE + c]     = sa[0];
        *(float4*)&As[0][0][r * LDS_STRIDE + c + 4] = sa[1];
        *(float4*)&As[0][1][r * LDS_STRIDE + c]     = sb[0];
        *(float4*)&As[0][1][r * LDS_STRIDE + c + 4] = sb[1];
        if (A2) {
            const float4* sc = (const float4*)(A2 + (size_t)(m0 + r) * HID + c);
            const float4* sd = (const float4*)(A2 + (size_t)r1 * HID + c);
            *(float4*)&As[1][0][r * LDS_STRIDE + c]     = sc[0];
            *(float4*)&As[1][0][r * LDS_STRIDE + c + 4] = sc[1];
            *(float4*)&As[1][1][r * LDS_STRIDE + c]     = sd[0];
            *(float4*)&As[1][1][r * LDS_STRIDE + c + 4] = sd[1];
        }
    }
    __syncthreads();

    v8f acc0 = {};
    v8f acc1 = {};
    const int nsets = A2 ? 2 : 1;
    for (int s = 0; s < nsets; ++s) {
        const float2* Bp = (const float2*)((s == 0) ? B1p : B2p);
        const float* At0 = As[s][0];
        const float* At1 = As[s][1];
        __builtin_prefetch(Bp, 0, 1);
        #pragma unroll 8
        for (int t = 0; t < KSTEPS; ++t) {
            const int kk = 4 * t + 2 * half;
            // B fragment: one 8B load -> { B[kk][n0+lrow], B[kk+1][n0+lrow] }
            v2f b  = *(const v2f*)&Bp[((size_t)t * HID + n0 + lrow) * 2 + half];
            v2f a0 = *(const v2f*)(At0 + lrow * LDS_STRIDE + kk);
            v2f a1 = *(const v2f*)(At1 + lrow * LDS_STRIDE + kk);
            acc0 = __builtin_amdgcn_wmma_f32_16x16x4_f32(
                false, a0, false, b, (short)0, acc0, false, false);
            acc1 = __builtin_amdgcn_wmma_f32_16x16x4_f32(
                false, a1, false, b, (short)0, acc1, false, false);
        }
    }

    const float bv = bias ? bias[n0 + lrow] : 0.f;
    #pragma unroll
    for (int j = 0; j < 8; ++j) {
        const int row0 = m0 + j + 8 * half;        // D layout: VGPR j -> rows j / j+8
        float v0 = acc0[j] + bv;
        if (relu) v0 = fmaxf(v0, 0.f);
        C[(size_t)row0 * HID + n0 + lrow] = v0;
        if (tile1_ok) {
            const int row1 = row0 + 16;
            float v1 = acc1[j] + bv;
            if (relu) v1 = fmaxf(v1, 0.f);
            C[(size_t)row1 * HID + n0 + lrow] = v1;
        }
    }
}

// one wave32 per edge: val = zs[src] + w*ww (float4/lane), atomic-max into keys[dst]
__global__ __launch_bounds__(256) void edge_scatter(
    const float* __restrict__ zs, const float* __restrict__ ew,
    const int* __restrict__ esrc, const int* __restrict__ edst,
    const float* __restrict__ ww, unsigned* __restrict__ mkeys,
    int* __restrict__ deg, int n_edges)
{
    const int wave = threadIdx.x >> 5;
    const int lane = threadIdx.x & 31;
    const int e = blockIdx.x * 8 + wave;
    if (e >= n_edges) return;

    const int   s = esrc[e];
    const int   d = edst[e];
    const float w = ew[e];

    float4 zv = ((const float4*)(zs + (size_t)s * HID))[lane];
    float4 wv = ((const float4*)ww)[lane];

    unsigned* mp = mkeys + (size_t)d * HID + lane * 4;
    atomicMax(mp + 0, fkey_enc(fmaf(w, wv.x, zv.x)));
    atomicMax(mp + 1, fkey_enc(fmaf(w, wv.y, zv.y)));
    atomicMax(mp + 2, fkey_enc(fmaf(w, wv.z, zv.z)));
    atomicMax(mp + 3, fkey_enc(fmaf(w, wv.w, zv.w)));
    if (lane == 0) atomicAdd(deg + d, 1);
}

// agg = deg>0 ? (zd' + decode(maxkey)) : 0   (in place over zd')
__global__ __launch_bounds__(256) void combine(
    float* __restrict__ zd, const unsigned* __restrict__ mkeys,
    const int* __restrict__ deg, int n_nodes)
{
    const int t  = blockIdx.x * blockDim.x + threadIdx.x;
    const int i  = t >> 5;
    const int c4 = (t & 31) * 4;
    if (i >= n_nodes) return;

    float4 out;
    if (deg[i] > 0) {
        float4 zv = *(const float4*)(zd + (size_t)i * HID + c4);
        uint4  k  = *(const uint4*)(mkeys + (size_t)i * HID + c4);
        out.x = zv.x + fkey_dec(k.x);
        out.y = zv.y + fkey_dec(k.y);
        out.z = zv.z + fkey_dec(k.z);
        out.w = zv.w + fkey_dec(k.w);
    } else {
        out = make_float4(0.f, 0.f, 0.f, 0.f);
    }
    *(float4*)(zd + (size_t)i * HID + c4) = out;
}

extern "C" void kernel_launch(void* const* d_in, const int* in_sizes, int n_in,
                              void* d_out, int out_size, void* d_ws, size_t ws_size,
                              hipStream_t stream) {
    const float* z    = (const float*)d_in[0];
    const float* ew   = (const float*)d_in[1];
    const int*   esrc = (const int*)d_in[2];
    const int*   edst = (const int*)d_in[3];
    const float* Wm   = (const float*)d_in[4];
    const float* bm   = (const float*)d_in[5];
    const float* W1   = (const float*)d_in[6];
    const float* b1   = (const float*)d_in[7];
    const float* W2   = (const float*)d_in[8];
    const float* b2   = (const float*)d_in[9];
    float* out = (float*)d_out;

    const int n_nodes = in_sizes[0] / HID;   // 50000
    const int n_edges = in_sizes[1];         // 800000
    const size_t node_elems = (size_t)n_nodes * HID;
    const size_t bpk = (size_t)KSTEPS * HID * 4;   // packed-B floats per matrix (16384)

    float*    zd    = (float*)d_ws;                 // -> agg (in place)
    float*    zs    = zd + node_elems;              // -> hidden (reused)
    unsigned* mkeys = (unsigned*)(zs + node_elems);
    int*      deg   = (int*)(mkeys + node_elems);   // contiguous after mkeys
    float*    BpWd  = (float*)(deg + n_nodes);
    float*    BpWs  = BpWd + bpk;
    float*    BpW1a = BpWs + bpk;
    float*    BpW1b = BpW1a + bpk;
    float*    BpW2  = BpW1b + bpk;

    const float* Wd  = Wm;
    const float* Ws_ = Wm + HID * HID;
    const float* ww  = Wm + 2 * HID * HID;
    const float* W1a = W1;
    const float* W1b = W1 + HID * HID;

    const int mblocks = (n_nodes + 31) / 32;   // 1563
    const int pblocks = (KSTEPS * HID + 255) / 256;  // 16

    // 0) repack all five 128x128 weight matrices into WMMA fragment order
    pack_b<<<pblocks, 256, 0, stream>>>(Wd,  BpWd);
    pack_b<<<pblocks, 256, 0, stream>>>(Ws_, BpWs);
    pack_b<<<pblocks, 256, 0, stream>>>(W1a, BpW1a);
    pack_b<<<pblocks, 256, 0, stream>>>(W1b, BpW1b);
    pack_b<<<pblocks, 256, 0, stream>>>(W2,  BpW2);

    // 1) reset max-keys + degree counters (graph-replay safe)
    const int nz = (int)node_elems + n_nodes;
    zero_u32<<<(nz + 255) / 256, 256, 0, stream>>>(mkeys, nz);

    // 2) per-node projections: zd' = z@Wd + bm ; zs = z@Ws
    gemm32<<<mblocks, 256, 0, stream>>>(z, BpWd, nullptr, nullptr, bm, zd, 0, n_nodes);
    gemm32<<<mblocks, 256, 0, stream>>>(z, BpWs, nullptr, nullptr, nullptr, zs, 0, n_nodes);

    // 3) edge scatter: elementwise max of (zs[src] + w*ww) into dst segments
    edge_scatter<<<(n_edges + 7) / 8, 256, 0, stream>>>(zs, ew, esrc, edst, ww,
                                                        mkeys, deg, n_edges);

    // 4) agg = deg>0 ? zd' + max : 0   (in place -> zd)
    combine<<<((size_t)n_nodes * 32 + 255) / 256, 256, 0, stream>>>(zd, mkeys, deg, n_nodes);

    // 5) hidden = relu(z@W1a + agg@W1b + b1)   (fused dual-A GEMM, into zs)
    gemm32<<<mblocks, 256, 0, stream>>>(z, BpW1a, zd, BpW1b, b1, zs, 1, n_nodes);

    // 6) out = hidden@W2 + b2
    gemm32<<<mblocks, 256, 0, stream>>>(zs, BpW2, nullptr, nullptr, b2, out, 0, n_nodes);
}